// upsample_maxindex_22565758173784
// MI455X (gfx1250) — compile-verified
//
#include <hip/hip_runtime.h>

// ---------------------------------------------------------------------------
// upsample_maxindex: y = zeros(163842,256); y[neigh[7i+max_index[i,f]], row(k)] = x[i,f]
// Pure memory scatter. MI455X strategy:
//   * output (167.8 MB) fits in 192 MB L2 -> zero with RT b128 stores, scatter RT
//   * streamed inputs (x, max_index) loaded with NT temporal hint to avoid
//     evicting the L2-resident output
//   * wave32, 4 elements/lane, global_load_b128 / global_store_b128 paths
// ---------------------------------------------------------------------------

#define FEAT 256

typedef __attribute__((ext_vector_type(4))) float v4f;
typedef __attribute__((ext_vector_type(4))) int   v4i;

// Zero-fill the output with wide stores (regular temporal hint: we WANT these
// lines resident in L2 so the scatter writes hit L2, not HBM).
__global__ void upsample_zero_kernel(v4f* __restrict__ out, long long n4) {
    long long i = (long long)blockIdx.x * blockDim.x + threadIdx.x;
    if (i < n4) {
        v4f z = (v4f)0.0f;
        out[i] = z;  // global_store_b128
    }
}

// One lane handles 4 consecutive features of one node:
//   k..k+3 share node i = k>>8, so one int4 + one float4 load feed 4 stores.
// row(k) = floor(k * step) in f64 matches numpy linspace+floor bit-exactly
// (step computed on host as 256.0/(N-1) in double, same as np.linspace).
__global__ void upsample_scatter_kernel(const float* __restrict__ x,
                                        const int*   __restrict__ max_index,
                                        const int*   __restrict__ neigh,
                                        float*       __restrict__ out,
                                        double step, long long ntot) {
    long long k = ((long long)blockIdx.x * blockDim.x + threadIdx.x) * 4;
    if (k >= ntot) return;

    int node = (int)(k >> 8);                 // k / FEAT
    const int* nbase = neigh + node * 7;      // 7-entry window, one hot line

    // Non-temporal streaming loads: single-use data, keep L2 for the output.
    v4i m  = __builtin_nontemporal_load((const v4i*)(max_index + k)); // b128 th:NT
    v4f xv = __builtin_nontemporal_load((const v4f*)(x + k));         // b128 th:NT

#pragma unroll
    for (int j = 0; j < 4; ++j) {
        int col = nbase[m[j]];                              // gather from LUT
        int row = (int)__builtin_floor((double)(k + j) * step);
        row = (row < FEAT - 1) ? row : (FEAT - 1);          // linspace endpoint clamp
        out[(long long)col * FEAT + row] = xv[j];           // scattered b32, RT -> L2
    }
}

extern "C" void kernel_launch(void* const* d_in, const int* in_sizes, int n_in,
                              void* d_out, int out_size, void* d_ws, size_t ws_size,
                              hipStream_t stream) {
    const float* x         = (const float*)d_in[0];   // (40962, 256) f32
    const int*   max_index = (const int*)d_in[1];     // (40962, 256) int
    const int*   neigh     = (const int*)d_in[2];     // (40962*7,)   int
    // d_in[3] = num_nodes scalar; output row count comes from out_size instead.

    long long ntot = (long long)in_sizes[0];          // raw_nodes * FEAT
    float*    out  = (float*)d_out;

    // Exactly replicate np.linspace's step computation (double precision).
    double step = (double)FEAT / (double)(ntot - 1);

    // Kernel 1: zero the whole output (out_size elements, multiple of 4).
    long long n4 = (long long)out_size / 4;
    int zb = (int)((n4 + 255) / 256);
    upsample_zero_kernel<<<zb, 256, 0, stream>>>((v4f*)out, n4);

    // Kernel 2: scatter (stream ordering guarantees zero-fill completes first).
    long long nthreads = ntot / 4;
    int sb = (int)((nthreads + 255) / 256);
    upsample_scatter_kernel<<<sb, 256, 0, stream>>>(x, max_index, neigh, out,
                                                    step, ntot);
}